// DeformableConv2d_44418551775254
// MI455X (gfx1250) — compile-verified
//
#include <hip/hip_runtime.h>

typedef __attribute__((ext_vector_type(2))) float v2f;
typedef __attribute__((ext_vector_type(8))) float v8f;

#define Bsz  4
#define Cch  64
#define Hd   128
#define Wd   128
#define Och  64
#define KKt  9
#define OFFC 18      // 2*K*K offset channels
#define NT   2       // N-tiles (of 16 positions) per deform block
#define SROW 68      // padded LDS row stride (floats): bank = (4n+c)%64

// patch geometry for WMMA offset conv: 3 rows x 66 cols x 64 ch
#define PROWS 3
#define PCOLS 66
#define PELEMS (PROWS * PCOLS * Cch)   // 12672

// ---------------------------------------------------------------------------
// Kernel 0: weight re-layouts (one shot).
//   wT   [kk][o(64)][c(64)]  <- weight[o,c,ky,kx]          (36864 el)
//   wOffT[kk][oc(32)][c(64)] <- offset_w[oc,c,ky,kx], oc>=18 zero (18432 el)
// Contiguous-c layout makes every WMMA A-fragment one global_load_b64.
// ---------------------------------------------------------------------------
__global__ __launch_bounds__(256)
void dcn_weight_xform(const float* __restrict__ weight,
                      const float* __restrict__ offw,
                      float* __restrict__ wT,
                      float* __restrict__ wOffT)
{
    int gid = blockIdx.x * 256 + threadIdx.x;
    if (gid < KKt * Och * Cch) {                   // main-conv weights
        int kk  = gid >> 12;
        int rem = gid & 4095;
        int o   = rem >> 6;
        int c   = rem & 63;
        wT[gid] = weight[(size_t)(o * Cch + c) * KKt + kk];
    } else {                                       // offset-conv weights (padded M=32)
        int g2  = gid - KKt * Och * Cch;
        if (g2 >= KKt * 32 * Cch) return;
        int kk  = g2 >> 11;                        // /2048
        int rem = g2 & 2047;
        int oc  = rem >> 6;
        int c   = rem & 63;
        wOffT[g2] = (oc < OFFC) ? offw[(size_t)(oc * Cch + c) * KKt + kk] : 0.f;
    }
}

// ---------------------------------------------------------------------------
// Kernel 1: offset conv via WMMA.  GEMM: M=32 (18 valid), K=576, N=B*H*W.
// Block = 128 threads (4 wave32) owns (b, one 64-wide row segment).
// LDS patch: x[b, :, h-1..h+1, wcol-1..wcol+64] stored [row][col][c] with
// padded stride 68 -> all 9 taps read shifted B-fragments, staged only once.
// Wave i computes N-tile i (16 positions) for both M-tiles.
// ---------------------------------------------------------------------------
__global__ __launch_bounds__(128)
void dcn_offset_wmma(const float* __restrict__ x,
                     const float* __restrict__ wOffT,
                     const float* __restrict__ offb,
                     float* __restrict__ off_out)
{
    __shared__ float patch[PROWS * PCOLS * SROW];  // 53.9 KB

    int b    = blockIdx.x >> 8;                    // 256 segments per batch
    int tile = blockIdx.x & 255;
    int h    = tile >> 1;                          // 2 segments per row (128/64)
    int wcol = (tile & 1) * 64;

    int lane = threadIdx.x & 31;
    int wave = threadIdx.x >> 5;

    // --- stage input patch (zero-padded at image borders) -------------------
    for (int idx = threadIdx.x; idx < PELEMS; idx += 128) {
        int c   = idx / (PROWS * PCOLS);           // 0..63
        int rem = idx - c * (PROWS * PCOLS);
        int row = rem / PCOLS;                     // 0..2
        int col = rem - row * PCOLS;               // 0..65
        int yg  = h - 1 + row;
        int xg  = wcol - 1 + col;
        float v = 0.f;
        if ((unsigned)yg < (unsigned)Hd && (unsigned)xg < (unsigned)Wd)
            v = x[((size_t)(b * Cch + c) * Hd + yg) * Wd + xg];
        patch[(row * PCOLS + col) * SROW + c] = v;
    }
    __syncthreads();

    v8f accA = {};                                 // oc 0..15
    v8f accB = {};                                 // oc 16..31 (only 16,17 valid)

    const int mn    = lane & 15;
    const int khalf = (lane >> 4) * 2;
    const int nbase = wave * 16;                   // this wave's N-tile

    for (int kk = 0; kk < KKt; ++kk) {
        int ky = kk / 3, kx = kk % 3;
        const float* wk = wOffT + (size_t)kk * (32 * Cch);
        // patch col for output n, tap kx: (wcol + n + kx - 1) - (wcol - 1) = n + kx
        const float* prow = &patch[(ky * PCOLS + nbase + kx + mn) * SROW + khalf];
#pragma unroll
        for (int c0 = 0; c0 < Cch; c0 += 4) {
            v2f a0 = *(const v2f*)&wk[(0  + mn) * Cch + c0 + khalf];
            v2f a1 = *(const v2f*)&wk[(16 + mn) * Cch + c0 + khalf];
            v2f bb = *(const v2f*)&prow[c0];
            accA = __builtin_amdgcn_wmma_f32_16x16x4_f32(
                false, a0, false, bb, (short)0, accA, false, false);
            accB = __builtin_amdgcn_wmma_f32_16x16x4_f32(
                false, a1, false, bb, (short)0, accB, false, false);
        }
    }

    // --- store: D layout: VGPR r -> M=r (lanes 0-15), M=r+8 (lanes 16-31) ---
    int n     = lane & 15;
    int mhalf = (lane >> 4) * 8;
#pragma unroll
    for (int r = 0; r < 8; ++r) {
        int oc = mhalf + r;                        // 0..15, all valid
        off_out[((size_t)(b * OFFC + oc) * Hd + h) * Wd + wcol + nbase + n]
            = accA[r] + offb[oc];
        int oc2 = 16 + mhalf + r;
        if (oc2 < OFFC)                            // only 16,17
            off_out[((size_t)(b * OFFC + oc2) * Hd + h) * Wd + wcol + nbase + n]
                = accB[r] + offb[oc2];
    }
}

// ---------------------------------------------------------------------------
// Kernel 2: deformable sampling + WMMA GEMM (unchanged from round 2).
// ---------------------------------------------------------------------------
__global__ __launch_bounds__(128)
void dcn_deform_wmma(const float* __restrict__ x,
                     const float* __restrict__ off,
                     const float* __restrict__ wT,
                     const float* __restrict__ bias,
                     float* __restrict__ out)
{
    __shared__ float sVal[NT * 16 * SROW];    // val tile [n][c], padded (8.7 KB)
    __shared__ int   sY0[NT * 16], sX0[NT * 16];
    __shared__ float sWy1[NT * 16], sWx1[NT * 16];

    int b    = blockIdx.x >> 9;               // 512 32-wide tiles per batch
    int tile = blockIdx.x & 511;
    int h    = tile >> 2;                     // 4 tiles per image row (128/32)
    int wcol = (tile & 3) * 32;

    int lane   = threadIdx.x & 31;
    int wave   = threadIdx.x >> 5;
    int o_base = wave * 16;

    v8f acc0 = {};
    v8f acc1 = {};

    const int mn    = lane & 15;
    const int khalf = (lane >> 4) * 2;

    for (int kk = 0; kk < KKt; ++kk) {
        if (threadIdx.x < NT * 16) {
            int n  = threadIdx.x;
            int w  = wcol + n;
            float oy = off[((size_t)(b * OFFC + 2 * kk)     * Hd + h) * Wd + w];
            float ox = off[((size_t)(b * OFFC + 2 * kk + 1) * Hd + h) * Wd + w];
            int ky = kk / 3, kx = kk % 3;
            float py = (float)(h - 1 + ky) + oy;
            float px = (float)(w - 1 + kx) + ox;
            float fy = floorf(py), fx = floorf(px);
            sY0[n]  = (int)fy;
            sX0[n]  = (int)fx;
            sWy1[n] = py - fy;
            sWx1[n] = px - fx;
        }
        __syncthreads();

#pragma unroll
        for (int i = 0; i < 16; ++i) {
            int idx = threadIdx.x + i * 128;
            int c = idx >> 5, n = idx & 31;
            int y0 = sY0[n], x0 = sX0[n];
            int y1 = y0 + 1, x1 = x0 + 1;
            float wy1 = sWy1[n], wx1 = sWx1[n];
            float wy0 = 1.f - wy1, wx0 = 1.f - wx1;
            const float* xc = x + ((size_t)(b * Cch + c) * Hd) * Wd;
            bool y0ok = (unsigned)y0 < (unsigned)Hd;
            bool y1ok = (unsigned)y1 < (unsigned)Hd;
            bool x0ok = (unsigned)x0 < (unsigned)Wd;
            bool x1ok = (unsigned)x1 < (unsigned)Wd;
            float v00 = (y0ok && x0ok) ? xc[y0 * Wd + x0] : 0.f;
            float v01 = (y0ok && x1ok) ? xc[y0 * Wd + x1] : 0.f;
            float v10 = (y1ok && x0ok) ? xc[y1 * Wd + x0] : 0.f;
            float v11 = (y1ok && x1ok) ? xc[y1 * Wd + x1] : 0.f;
            sVal[n * SROW + c] = fmaf(wy0 * wx0, v00,
                                 fmaf(wy0 * wx1, v01,
                                 fmaf(wy1 * wx0, v10, wy1 * wx1 * v11)));
        }
        __syncthreads();

        const float* wk = wT + (size_t)kk * (Och * Cch);
#pragma unroll
        for (int c0 = 0; c0 < Cch; c0 += 4) {
            v2f a   = *(const v2f*)&wk[(o_base + mn) * Cch + c0 + khalf];
            v2f bb0 = *(const v2f*)&sVal[mn * SROW + c0 + khalf];
            v2f bb1 = *(const v2f*)&sVal[(16 + mn) * SROW + c0 + khalf];
            acc0 = __builtin_amdgcn_wmma_f32_16x16x4_f32(
                false, a, false, bb0, (short)0, acc0, false, false);
            acc1 = __builtin_amdgcn_wmma_f32_16x16x4_f32(
                false, a, false, bb1, (short)0, acc1, false, false);
        }
        __syncthreads();
    }

    int n     = lane & 15;
    int mhalf = (lane >> 4) * 8;
#pragma unroll
    for (int r = 0; r < 8; ++r) {
        int o = o_base + mhalf + r;
        float bo = bias[o];
        size_t base = ((size_t)(b * Och + o) * Hd + h) * Wd + wcol + n;
        out[base]      = acc0[r] + bo;
        out[base + 16] = acc1[r] + bo;
    }
}

// ---------------------------------------------------------------------------
extern "C" void kernel_launch(void* const* d_in, const int* in_sizes, int n_in,
                              void* d_out, int out_size, void* d_ws, size_t ws_size,
                              hipStream_t stream) {
    const float* x      = (const float*)d_in[0];   // [4,64,128,128]
    const float* offw   = (const float*)d_in[1];   // [18,64,3,3]
    const float* offb   = (const float*)d_in[2];   // [18]
    const float* weight = (const float*)d_in[3];   // [64,64,3,3]
    const float* bias   = (const float*)d_in[4];   // [64]
    float*       out    = (float*)d_out;           // [4,64,128,128]

    float* off_buf = (float*)d_ws;                             // [4,18,128,128] = 4.5 MB
    float* wT      = off_buf + (size_t)Bsz * OFFC * Hd * Wd;   // [9,64,64]
    float* wOffT   = wT + (size_t)KKt * Och * Cch;             // [9,32,64]

    // Kernel 0: weight re-layouts (36864 + 18432 = 55296 threads)
    dcn_weight_xform<<<(KKt * Och * Cch + KKt * 32 * Cch) / 256, 256, 0, stream>>>(
        weight, offw, wT, wOffT);

    // Kernel 1: offset conv via WMMA (B * H * W/64 = 1024 blocks)
    dcn_offset_wmma<<<Bsz * Hd * (Wd / 64), 128, 0, stream>>>(x, wOffT, offb, off_buf);

    // Kernel 2: deformable gather + WMMA GEMM (B * H*W/32 = 2048 blocks)
    dcn_deform_wmma<<<Bsz * (Hd * Wd / 32), 128, 0, stream>>>(x, off_buf, wT, bias, out);
}